// GGNN_34729105555600
// MI455X (gfx1250) — compile-verified
//
#include <hip/hip_runtime.h>
#include <hip/hip_bf16.h>
#include <stdint.h>

// ---------------------------------------------------------------------------
// GatedGraphConv (GGNN): 3 x [dense GEMM -> edge gather/scatter-add -> GRU],
// then MLP + log_softmax.  All f32.  CDNA5 paths used:
//   * V_WMMA_F32_16X16X4_F32 for all large GEMMs (f32-exact vs reference)
//   * GLOBAL_LOAD_ASYNC_TO_LDS_B128 + S_WAIT_ASYNCCNT to stage A-tiles in LDS
//   * GLOBAL_PREFETCH via __builtin_prefetch in the scatter kernel
// ---------------------------------------------------------------------------

#define GG_N 50000
#define GG_E 600000
#define GG_C 128
#define GG_L 3
#define GG_HID 32
#define GG_CLS 7

#define LDA 130   // padded LDS row stride (floats) for 16x128 A tiles
#define LDG 388   // padded LDS row stride (floats) for 16x384 gate tiles

typedef float v2f __attribute__((ext_vector_type(2)));
typedef float v8f __attribute__((ext_vector_type(8)));

// D = A(16x4) * B(4x16) + C, f32 WMMA.
__device__ __forceinline__ v8f wmma_f32_4(v2f a, v2f b, v8f c) {
  return __builtin_amdgcn_wmma_f32_16x16x4_f32(false, a, false, b, (short)0, c,
                                               false, false);
}

// LDS byte offset of a shared-memory pointer: flat shared-aperture addresses
// carry the LDS offset in addr[31:0].
__device__ __forceinline__ uint32_t lds_off(const void* p) {
  return (uint32_t)(uintptr_t)p;
}

// Async copy 16B/lane global -> LDS (tracked by ASYNCcnt).
__device__ __forceinline__ void async_ld_b128(uint32_t lds, const void* g) {
  asm volatile("global_load_async_to_lds_b128 %0, %1, off"
               :
               : "v"(lds), "v"((uint64_t)(uintptr_t)g)
               : "memory");
}

__device__ __forceinline__ void wait_async0() {
  asm volatile("s_wait_asynccnt 0" ::: "memory");
}

// ---------------------------------------------------------------------------
__global__ void ggnn_copy(const float* __restrict__ src, float* __restrict__ dst,
                          int n) {
  int i = blockIdx.x * blockDim.x + threadIdx.x;
  if (i < n) dst[i] = src[i];
}

__global__ void ggnn_zero(float* __restrict__ p, int n) {
  int i = blockIdx.x * blockDim.x + threadIdx.x;
  if (i < n) p[i] = 0.0f;
}

// WT[j*128+k] = W[k*128+j]  (so WMMA B-frags read 2 consecutive floats)
__global__ void ggnn_transpose128(const float* __restrict__ W,
                                  float* __restrict__ WT) {
  int i = blockIdx.x * blockDim.x + threadIdx.x;  // 0..16383
  int r = i >> 7, c = i & 127;
  WT[c * GG_C + r] = W[r * GG_C + c];
}

// ---------------------------------------------------------------------------
// m = h @ W : (N x 128) @ (128 x 128), using WT[j][k].
// Block = 256 thr = 8 waves; block owns a 16-row tile, wave w owns cols
// [16w,16w+16).  A tile staged once per block via async-to-LDS.
// ---------------------------------------------------------------------------
__global__ void __launch_bounds__(256)
ggnn_gemm_m(const float* __restrict__ H, const float* __restrict__ WT,
            float* __restrict__ M) {
  __shared__ float hT[16 * LDA];
  const int wave = threadIdx.x >> 5;
  const int lane = threadIdx.x & 31;
  const int row0 = blockIdx.x * 16;

  {  // stage 16x128 tile: 16 threads/row, 8 floats (2 x b128) each
    const int r  = threadIdx.x >> 4;
    const int c0 = (threadIdx.x & 15) * 8;
    const float* g   = H + (row0 + r) * GG_C + c0;
    const uint32_t l = lds_off(&hT[r * LDA + c0]);
    async_ld_b128(l, g);
    async_ld_b128(l + 16, g + 4);
  }
  wait_async0();
  __syncthreads();

  const int col0 = wave * 16;
  const int mn   = lane & 15;        // M for A-frag, N for B-frag
  const int koff = (lane >> 4) * 2;  // K sub-offset for this lane half

  v8f acc = {};
#pragma unroll 4
  for (int k0 = 0; k0 < GG_C; k0 += 4) {
    const float* ap = &hT[mn * LDA + k0 + koff];
    v2f a;
    a[0] = ap[0];
    a[1] = ap[1];
    const float* bp = WT + (col0 + mn) * GG_C + (k0 + koff);
    v2f b;
    b[0] = bp[0];
    b[1] = bp[1];
    acc = wmma_f32_4(a, b, acc);
  }
  const int rhi = (lane >> 4) * 8;
#pragma unroll
  for (int i = 0; i < 8; ++i)
    M[(row0 + rhi + i) * GG_C + col0 + mn] = acc[i];
}

// ---------------------------------------------------------------------------
// Edge scatter: agg[dst] += ew[e] * m[src].  One edge per wave: 32 lanes x
// float4 -> one contiguous 512B gather + one contiguous 512B atomic region.
// ---------------------------------------------------------------------------
__global__ void __launch_bounds__(256)
ggnn_scatter(const float* __restrict__ M, const int* __restrict__ src,
             const int* __restrict__ dst, const float* __restrict__ ew,
             float* __restrict__ agg) {
  const long long tid = (long long)blockIdx.x * blockDim.x + threadIdx.x;
  const int e = (int)(tid >> 5);
  if (e >= GG_E) return;
  const int q = (int)(tid & 31) * 4;
  const int s = src[e];
  const int d = dst[e];
  const float w = ew[e];
  if (e + 8 < GG_E)  // speculative prefetch of an upcoming source row
    __builtin_prefetch(M + (long long)src[e + 8] * GG_C + q, 0, 1);
  const float4 mv = *(const float4*)(M + (long long)s * GG_C + q);
  float* ap = agg + (long long)d * GG_C + q;
  atomicAdd(ap + 0, w * mv.x);
  atomicAdd(ap + 1, w * mv.y);
  atomicAdd(ap + 2, w * mv.z);
  atomicAdd(ap + 3, w * mv.w);
}

// ---------------------------------------------------------------------------
// Fused GRU cell for one 16-node tile.
//   gx = agg @ w_ih^T   (16x128)@(128x384)     gh = h @ w_hh^T
// agg/h tiles staged via async-to-LDS; gate pre-activations accumulate in LDS
// (stride padded for conflict-free accumulator spill); elementwise gates then
// update h in place (block owns its rows exclusively).
// ---------------------------------------------------------------------------
__global__ void __launch_bounds__(256)
ggnn_gru(const float* __restrict__ AGG, float* __restrict__ H,
         const float* __restrict__ w_ih, const float* __restrict__ w_hh,
         const float* __restrict__ b_ih, const float* __restrict__ b_hh) {
  __shared__ float gx[16 * LDG];
  __shared__ float gh[16 * LDG];
  __shared__ float aT[16 * LDA];
  __shared__ float hT[16 * LDA];

  const int wave = threadIdx.x >> 5;
  const int lane = threadIdx.x & 31;
  const int row0 = blockIdx.x * 16;

  {  // stage agg & h tiles
    const int r  = threadIdx.x >> 4;
    const int c0 = (threadIdx.x & 15) * 8;
    const float* ga   = AGG + (row0 + r) * GG_C + c0;
    const float* gh_g = H + (row0 + r) * GG_C + c0;
    const uint32_t la = lds_off(&aT[r * LDA + c0]);
    const uint32_t lh = lds_off(&hT[r * LDA + c0]);
    async_ld_b128(la, ga);
    async_ld_b128(la + 16, ga + 4);
    async_ld_b128(lh, gh_g);
    async_ld_b128(lh + 16, gh_g + 4);
  }
  wait_async0();
  __syncthreads();

  const int mn   = lane & 15;
  const int koff = (lane >> 4) * 2;
  const int rhi  = (lane >> 4) * 8;

  for (int t = 0; t < 3; ++t) {
    const int col0 = (wave * 3 + t) * 16;  // 0..368
    v8f accx = {};
    v8f acch = {};
#pragma unroll 4
    for (int k0 = 0; k0 < GG_C; k0 += 4) {
      const float* ax = &aT[mn * LDA + k0 + koff];
      v2f a;
      a[0] = ax[0];
      a[1] = ax[1];
      const float* bx = w_ih + (col0 + mn) * GG_C + (k0 + koff);
      v2f b;
      b[0] = bx[0];
      b[1] = bx[1];
      accx = wmma_f32_4(a, b, accx);

      const float* ah = &hT[mn * LDA + k0 + koff];
      v2f a2;
      a2[0] = ah[0];
      a2[1] = ah[1];
      const float* bh = w_hh + (col0 + mn) * GG_C + (k0 + koff);
      v2f b2;
      b2[0] = bh[0];
      b2[1] = bh[1];
      acch = wmma_f32_4(a2, b2, acch);
    }
#pragma unroll
    for (int i = 0; i < 8; ++i) {
      gx[(rhi + i) * LDG + col0 + mn] = accx[i];
      gh[(rhi + i) * LDG + col0 + mn] = acch[i];
    }
  }
  __syncthreads();

  // gates: 16 nodes x 128 channels, torch GRUCell order (r, z, n)
  for (int idx = threadIdx.x; idx < 16 * GG_C; idx += 256) {
    const int n = idx >> 7;
    const int c = idx & (GG_C - 1);
    const float xr = gx[n * LDG + c] + b_ih[c];
    const float xz = gx[n * LDG + 128 + c] + b_ih[128 + c];
    const float xn = gx[n * LDG + 256 + c] + b_ih[256 + c];
    const float hr = gh[n * LDG + c] + b_hh[c];
    const float hz = gh[n * LDG + 128 + c] + b_hh[128 + c];
    const float hn = gh[n * LDG + 256 + c] + b_hh[256 + c];
    const float r = 1.0f / (1.0f + expf(-(xr + hr)));
    const float z = 1.0f / (1.0f + expf(-(xz + hz)));
    const float nn = tanhf(xn + r * hn);
    const float hold = hT[n * LDA + c];
    H[(row0 + n) * GG_C + c] = (1.0f - z) * nn + z * hold;
  }
}

// ---------------------------------------------------------------------------
// Fused MLP head: 128 -> 32 -> 32 -> 32 -> 7 (+log_softmax).  One wave/node.
// ---------------------------------------------------------------------------
__global__ void __launch_bounds__(256)
ggnn_mlp(const float* __restrict__ H, const float* __restrict__ w0,
         const float* __restrict__ b0, const float* __restrict__ w1,
         const float* __restrict__ b1, const float* __restrict__ w2,
         const float* __restrict__ b2, const float* __restrict__ w3,
         const float* __restrict__ b3, float* __restrict__ out) {
  __shared__ float bufA[8][GG_HID];
  __shared__ float bufB[8][GG_HID];
  __shared__ float lg[8][8];

  const int wv = threadIdx.x >> 5;
  const int j  = threadIdx.x & 31;
  const int n  = blockIdx.x * 8 + wv;
  const float* h = H + (long long)n * GG_C;

  float acc = b0[j];
#pragma unroll 8
  for (int k = 0; k < GG_C; ++k) acc += h[k] * w0[j * GG_C + k];
  bufA[wv][j] = tanhf(acc);
  __syncthreads();

  acc = b1[j];
#pragma unroll
  for (int k = 0; k < GG_HID; ++k) acc += bufA[wv][k] * w1[j * GG_HID + k];
  bufB[wv][j] = tanhf(acc);
  __syncthreads();

  acc = b2[j];
#pragma unroll
  for (int k = 0; k < GG_HID; ++k) acc += bufB[wv][k] * w2[j * GG_HID + k];
  bufA[wv][j] = tanhf(acc);
  __syncthreads();

  float logit = 0.0f;
  if (j < GG_CLS) {
    logit = b3[j];
#pragma unroll
    for (int k = 0; k < GG_HID; ++k) logit += bufA[wv][k] * w3[j * GG_HID + k];
    lg[wv][j] = logit;
  }
  __syncthreads();

  if (j < GG_CLS) {
    float mx = lg[wv][0];
#pragma unroll
    for (int k = 1; k < GG_CLS; ++k) mx = fmaxf(mx, lg[wv][k]);
    float s = 0.0f;
#pragma unroll
    for (int k = 0; k < GG_CLS; ++k) s += expf(lg[wv][k] - mx);
    out[(long long)n * GG_CLS + j] = logit - mx - logf(s);
  }
}

// ---------------------------------------------------------------------------
extern "C" void kernel_launch(void* const* d_in, const int* in_sizes, int n_in,
                              void* d_out, int out_size, void* d_ws,
                              size_t ws_size, hipStream_t stream) {
  (void)in_sizes; (void)n_in; (void)out_size; (void)ws_size;
  const float* x    = (const float*)d_in[0];
  const int*   ei   = (const int*)d_in[1];   // [2, E] flat: src then dst
  const float* ew   = (const float*)d_in[2];
  const float* cw   = (const float*)d_in[3]; // [L, C, C]
  const float* w_ih = (const float*)d_in[4];
  const float* w_hh = (const float*)d_in[5];
  const float* b_ih = (const float*)d_in[6];
  const float* b_hh = (const float*)d_in[7];
  const float* w0 = (const float*)d_in[8];
  const float* b0 = (const float*)d_in[9];
  const float* w1 = (const float*)d_in[10];
  const float* b1 = (const float*)d_in[11];
  const float* w2 = (const float*)d_in[12];
  const float* b2 = (const float*)d_in[13];
  const float* w3 = (const float*)d_in[14];
  const float* b3 = (const float*)d_in[15];

  float* h   = (float*)d_ws;                 // N*C
  float* m   = h + (size_t)GG_N * GG_C;      // N*C
  float* agg = m + (size_t)GG_N * GG_C;      // N*C
  float* wt  = agg + (size_t)GG_N * GG_C;    // C*C (transposed conv weight)

  const int NC = GG_N * GG_C;
  const int rowTiles = GG_N / 16;            // 3125, exact

  ggnn_copy<<<(NC + 255) / 256, 256, 0, stream>>>(x, h, NC);

  for (int l = 0; l < GG_L; ++l) {
    ggnn_transpose128<<<(GG_C * GG_C) / 256, 256, 0, stream>>>(
        cw + (size_t)l * GG_C * GG_C, wt);
    ggnn_gemm_m<<<rowTiles, 256, 0, stream>>>(h, wt, m);
    ggnn_zero<<<(NC + 255) / 256, 256, 0, stream>>>(agg, NC);
    const long long sthreads = (long long)GG_E * 32;
    ggnn_scatter<<<(int)((sthreads + 255) / 256), 256, 0, stream>>>(
        m, ei, ei + GG_E, ew, agg);
    ggnn_gru<<<rowTiles, 256, 0, stream>>>(agg, h, w_ih, w_hh, b_ih, b_hh);
  }

  ggnn_mlp<<<GG_N / 8, 256, 0, stream>>>(h, w0, b0, w1, b1, w2, b2, w3, b3,
                                         (float*)d_out);
}